// CumsumEmbedding_46935402611133
// MI455X (gfx1250) — compile-verified
//
#include <hip/hip_runtime.h>

// ---------------------------------------------------------------------------
// CumsumEmbedding for MI455X (gfx1250, wave32)
//   Phase 1 (tiny, ~12 MB traffic): hierarchical column prefix-sum over the
//     8192x128 embedding matrix + fused LayerNorm, using V_WMMA_F32_16X16X4_F32
//     (scan-as-matmul with a lower-triangular ones matrix) -> 4 MB table in ws.
//   Phase 2 (dominant, 256 MB stream): gather rows by index with B128 loads
//     (table stays hot in L2) and non-temporal B128 stores for the output.
// ---------------------------------------------------------------------------

#define N_EMB   8192
#define EDIM    128
#define CHUNK   128                 // rows per scan chunk
#define NCHUNK  (N_EMB / CHUNK)     // 64
#define LN_EPS  1e-5f

typedef __attribute__((ext_vector_type(2))) float v2f;
typedef __attribute__((ext_vector_type(4))) float v4f;
typedef __attribute__((ext_vector_type(8))) float v8f;

// Kernel 1: per-chunk column sums. 64 blocks x 128 threads (thread = column).
__global__ void cse_partials(const float* __restrict__ E, float* __restrict__ partial) {
    const int col = threadIdx.x;
    const int c   = blockIdx.x;
    const float* p = E + (size_t)c * CHUNK * EDIM + col;
    float acc = 0.0f;
    #pragma unroll 8
    for (int r = 0; r < CHUNK; ++r) acc += p[(size_t)r * EDIM];
    partial[c * EDIM + col] = acc;
}

// Kernel 2: exclusive scan over the 64 chunk partials, per column.
__global__ void cse_scan_partials(const float* __restrict__ partial, float* __restrict__ offs) {
    const int col = threadIdx.x;
    float run = 0.0f;
    for (int c = 0; c < NCHUNK; ++c) {
        float p = partial[c * EDIM + col];
        offs[c * EDIM + col] = run;
        run += p;
    }
}

// Kernel 3: WMMA tile scan + fused LayerNorm. 64 blocks x 1 wave (32 lanes).
// Each wave scans its 128-row chunk: 8 row-tiles x 8 col-tiles, each tile is
// L(16x16, lower-tri ones) @ X(16x16) done as 4 chained 16x16x4 f32 WMMAs.
__global__ void __launch_bounds__(32)
cse_scan_ln(const float* __restrict__ E,
            const float* __restrict__ offs,
            const float* __restrict__ gamma,
            const float* __restrict__ beta,
            float* __restrict__ normed) {
    const int chunk = blockIdx.x;
    const int lane  = threadIdx.x;
    const int half  = lane >> 4;     // lane group: 0 = lanes 0-15, 1 = lanes 16-31
    const int ln    = lane & 15;

    // A = lower-triangular ones, 16x16, as four 16x4 K-slices.
    // A layout (32-bit 16x4): M = lane%16; VGPR0 -> K = 2*half, VGPR1 -> K = 2*half+1.
    v2f a[4];
    #pragma unroll
    for (int k = 0; k < 4; ++k) {
        const int k0 = k * 4 + 2 * half;
        a[k].x = (k0     <= ln) ? 1.0f : 0.0f;
        a[k].y = (k0 + 1 <= ln) ? 1.0f : 0.0f;
    }

    float carry[8], g[8], bt[8];
    #pragma unroll
    for (int t = 0; t < 8; ++t) {
        carry[t] = offs[chunk * EDIM + t * 16 + ln];  // exclusive prefix for this chunk
        g[t]  = gamma[t * 16 + ln];
        bt[t] = beta [t * 16 + ln];
    }

    for (int rt = 0; rt < 8; ++rt) {
        const int R = chunk * CHUNK + rt * 16;        // first row of this 16-row tile
        v8f dacc[8];                                  // 8 col-tiles of the 16x128 cumsum slab

        #pragma unroll
        for (int t = 0; t < 8; ++t) {
            // C initialized with the running column carry (broadcast down all rows).
            v8f c;
            #pragma unroll
            for (int v = 0; v < 8; ++v) c[v] = carry[t];
            #pragma unroll
            for (int k = 0; k < 4; ++k) {
                // B layout (32-bit 4x16): N = lane%16; VGPR0 -> K = 2*half, VGPR1 -> K = 2*half+1
                const int row = R + k * 4 + 2 * half;
                v2f b;
                b.x = E[(size_t)row       * EDIM + t * 16 + ln];
                b.y = E[(size_t)(row + 1) * EDIM + t * 16 + ln];
                c = __builtin_amdgcn_wmma_f32_16x16x4_f32(
                        /*neg_a=*/false, a[k], /*neg_b=*/false, b,
                        /*c_mod=*/(short)0, c, /*reuse_a=*/false, /*reuse_b=*/false);
            }
            dacc[t] = c;
            // New carry = inclusive cumsum at row R+15 = D element (M=15,N) = VGPR7, lane 16+N.
            carry[t] = __shfl(c[7], 16 + ln, 32);
        }

        // Fused LayerNorm. D layout: lane l, VGPR v holds row M = v + 8*half, col = t*16 + ln.
        // A row's 128 values live on the 16 lanes of one half -> xor-shuffle reduce (1,2,4,8).
        #pragma unroll
        for (int v = 0; v < 8; ++v) {
            float s = 0.0f;
            #pragma unroll
            for (int t = 0; t < 8; ++t) s += dacc[t][v];
            #pragma unroll
            for (int m = 8; m >= 1; m >>= 1) s += __shfl_xor(s, m, 32);
            const float mu = s * (1.0f / 128.0f);

            float cs = 0.0f;
            #pragma unroll
            for (int t = 0; t < 8; ++t) { const float xc = dacc[t][v] - mu; cs += xc * xc; }
            #pragma unroll
            for (int m = 8; m >= 1; m >>= 1) cs += __shfl_xor(cs, m, 32);
            const float rs = rsqrtf(cs * (1.0f / 128.0f) + LN_EPS);

            const int rowM = R + v + 8 * half;
            #pragma unroll
            for (int t = 0; t < 8; ++t)
                normed[(size_t)rowM * EDIM + t * 16 + ln] = (dacc[t][v] - mu) * rs * g[t] + bt[t];
        }
    }
}

// Kernel 4: gather. One wave per output row: 32 lanes x 16 B = 512 B/row.
// Table loads hit L2 (4 MB, ~64x reuse); output stores are non-temporal B128.
__global__ void cse_gather(const int* __restrict__ idx,
                           const float* __restrict__ normed,
                           float* __restrict__ out, int nrows) {
    const int wave = (int)((blockIdx.x * blockDim.x + threadIdx.x) >> 5);
    const int lane = threadIdx.x & 31;
    if (wave >= nrows) return;
    const int row = idx[wave];
    const v4f* __restrict__ src = (const v4f*)(normed + (size_t)row * EDIM);
    v4f val = src[lane];
    v4f* dst = (v4f*)(out + (size_t)wave * EDIM);
    __builtin_nontemporal_store(val, dst + lane);
}

extern "C" void kernel_launch(void* const* d_in, const int* in_sizes, int n_in,
                              void* d_out, int out_size, void* d_ws, size_t ws_size,
                              hipStream_t stream) {
    const int*   idx   = (const int*)  d_in[0];   // [64, 8192] int32
    const float* E     = (const float*)d_in[1];   // [8192, 128] f32
    const float* gamma = (const float*)d_in[2];   // [128] f32
    const float* beta  = (const float*)d_in[3];   // [128] f32
    float* out = (float*)d_out;                   // [64, 8192, 128] f32

    // Workspace layout: normed table (4 MB) | chunk partials (32 KB) | exclusive offsets (32 KB)
    float* ws_normed  = (float*)d_ws;
    float* ws_partial = ws_normed  + (size_t)N_EMB * EDIM;
    float* ws_offs    = ws_partial + (size_t)NCHUNK * EDIM;

    cse_partials     <<<NCHUNK, EDIM, 0, stream>>>(E, ws_partial);
    cse_scan_partials<<<1,      EDIM, 0, stream>>>(ws_partial, ws_offs);
    cse_scan_ln      <<<NCHUNK, 32,   0, stream>>>(E, ws_offs, gamma, beta, ws_normed);

    const int nrows = in_sizes[0];                // 524288 output rows
    const int rows_per_block = 256 / 32;
    const int blocks = (nrows + rows_per_block - 1) / rows_per_block;
    cse_gather<<<blocks, 256, 0, stream>>>(idx, ws_normed, out, nrows);
}